// GAT_48850958025269
// MI455X (gfx1250) — compile-verified
//
#include <hip/hip_runtime.h>
#include <hip/hip_bf16.h>

#define NEG_SLOPE 0.2f

typedef __attribute__((ext_vector_type(2))) float v2f;
typedef __attribute__((ext_vector_type(8))) float v8f;

__device__ __forceinline__ float lrelu(float x) { return x > 0.0f ? x : NEG_SLOPE * x; }

// order-preserving float<->uint map for atomicMax on floats
__device__ __forceinline__ unsigned fenc(float x) {
  unsigned u = __float_as_uint(x);
  return (u & 0x80000000u) ? ~u : (u | 0x80000000u);
}
__device__ __forceinline__ float fdec(unsigned k) {
  unsigned u = (k & 0x80000000u) ? (k & 0x7FFFFFFFu) : ~k;
  return __uint_as_float(u);
}

__device__ __forceinline__ void edge_sd(const int* __restrict__ ei, int E, int e, int& s, int& d) {
  if (e < E) { s = ei[e]; d = ei[E + e]; }
  else       { s = e - E; d = e - E; }        // self-loops appended
}

// ---------------- WMMA fp32 GEMM: C[N,Kout] = A[N,K] @ B[K,Kout] ----------------
// One wave -> 16(M) x 64(N) output strip (4 accumulators), K-loop step 4.
__global__ void gemm_wmma_f32(const float* __restrict__ A, const float* __restrict__ B,
                              float* __restrict__ C, int N, int K, int Kout) {
  const int lane = threadIdx.x & 31;
  const int wave = (int)((blockIdx.x * blockDim.x + threadIdx.x) >> 5);
  const int groupsN = Kout >> 6;                 // groups of 4 16-wide tiles
  const int tm = wave / groupsN;
  const int tg = wave % groupsN;
  if (tm * 16 >= N) return;                      // wave-uniform exit (EXEC all-1 inside)

  const int rowA  = tm * 16 + (lane & 15);
  const int kh    = (lane >> 4) << 1;            // lanes 16-31 hold K = k0+2, k0+3
  const int colB0 = tg * 64 + (lane & 15);

  v8f acc0 = {}, acc1 = {}, acc2 = {}, acc3 = {};
  const float* Arow = A + (size_t)rowA * K;

  for (int k0 = 0; k0 < K; k0 += 4) {
    const int kb = k0 + kh;
    v2f a = { Arow[kb], Arow[kb + 1] };
    const float* B0 = B + (size_t)kb * Kout + colB0;
    const float* B1 = B0 + Kout;
    v2f b0 = { B0[0],  B1[0]  };
    v2f b1 = { B0[16], B1[16] };
    v2f b2 = { B0[32], B1[32] };
    v2f b3 = { B0[48], B1[48] };
    acc0 = __builtin_amdgcn_wmma_f32_16x16x4_f32(false, a, false, b0, (short)0, acc0, false, false);
    acc1 = __builtin_amdgcn_wmma_f32_16x16x4_f32(false, a, false, b1, (short)0, acc1, false, false);
    acc2 = __builtin_amdgcn_wmma_f32_16x16x4_f32(false, a, false, b2, (short)0, acc2, false, false);
    acc3 = __builtin_amdgcn_wmma_f32_16x16x4_f32(false, a, false, b3, (short)0, acc3, false, false);
  }

  // C/D layout: lane[0..15] -> rows M=v, lane[16..31] -> rows M=v+8; col = lane&15
  const int rowBase = tm * 16 + ((lane >> 4) << 3);
  float* Cr = C + (size_t)rowBase * Kout + tg * 64 + (lane & 15);
#pragma unroll
  for (int v = 0; v < 8; ++v) {
    Cr[(size_t)v * Kout + 0]  = acc0[v];
    Cr[(size_t)v * Kout + 16] = acc1[v];
    Cr[(size_t)v * Kout + 32] = acc2[v];
    Cr[(size_t)v * Kout + 48] = acc3[v];
  }
}

// ---------------- attention logits: alS/alD[n,h] = dot(h[n,h,:], a_{src,dst}[h,:]) ----
__global__ void attn_logits(const float* __restrict__ Hf, const float* __restrict__ a_src,
                            const float* __restrict__ a_dst, float* __restrict__ alS,
                            float* __restrict__ alD, int N, int Hh, int C) {
  const int lane = threadIdx.x & 31;
  const int wave = (int)((blockIdx.x * blockDim.x + threadIdx.x) >> 5);
  const int node = wave / Hh, head = wave % Hh;
  if (node >= N) return;
  const float* hv = Hf + (size_t)node * Hh * C + head * C;
  const float* as = a_src + head * C;
  const float* ad = a_dst + head * C;
  float ss = 0.0f, sd = 0.0f;
  for (int c = lane; c < C; c += 32) {
    float v = hv[c];
    ss += v * as[c];
    sd += v * ad[c];
  }
#pragma unroll
  for (int off = 16; off; off >>= 1) {
    ss += __shfl_down(ss, off, 32);
    sd += __shfl_down(sd, off, 32);
  }
  if (lane == 0) { alS[node * Hh + head] = ss; alD[node * Hh + head] = sd; }
}

// ---------------- fills ----------------
__global__ void fill_f32(float* p, float v, int n) {
  int i = blockIdx.x * blockDim.x + threadIdx.x;
  if (i < n) p[i] = v;
}
__global__ void fill_u32(unsigned* p, unsigned v, int n) {
  int i = blockIdx.x * blockDim.x + threadIdx.x;
  if (i < n) p[i] = v;
}

// ---------------- edge pass 1: segment max of leaky_relu(alS[src]+alD[dst]) over dst --
__global__ void edge_max(const int* __restrict__ ei, int E, int Etot,
                         const float* __restrict__ alS, const float* __restrict__ alD,
                         unsigned* __restrict__ mEnc, int Hh) {
  int idx = blockIdx.x * blockDim.x + threadIdx.x;
  if (idx >= Etot * Hh) return;
  int e = idx / Hh, h = idx % Hh;
  int s, d; edge_sd(ei, E, e, s, d);
  float v = lrelu(alS[s * Hh + h] + alD[d * Hh + h]);
  atomicMax(&mEnc[d * Hh + h], fenc(v));
}

// ---------------- edge pass 2: p = exp(e - m[dst]); s[dst] += p --------------------
__global__ void edge_exp(const int* __restrict__ ei, int E, int Etot,
                         const float* __restrict__ alS, const float* __restrict__ alD,
                         const unsigned* __restrict__ mEnc, float* __restrict__ pB,
                         float* __restrict__ sSum, int Hh) {
  int idx = blockIdx.x * blockDim.x + threadIdx.x;
  if (idx >= Etot * Hh) return;
  int e = idx / Hh, h = idx % Hh;
  int s, d; edge_sd(ei, E, e, s, d);
  float v = lrelu(alS[s * Hh + h] + alD[d * Hh + h]);
  float p = __expf(v - fdec(mEnc[d * Hh + h]));
  pB[idx] = p;
  atomicAdd(&sSum[d * Hh + h], p);
}

// ---------------- edge pass 3: out[dst] += alpha * h[src] (one wave per edge) ------
__global__ void edge_msg(const int* __restrict__ ei, int E, int Etot,
                         const float* __restrict__ pB, const float* __restrict__ sSum,
                         const float* __restrict__ Hf, float* __restrict__ outF,
                         int Hh, int C) {
  const int lane = threadIdx.x & 31;
  const int wave = (int)((blockIdx.x * blockDim.x + threadIdx.x) >> 5);
  if (wave >= Etot) return;
  int s, d; edge_sd(ei, E, wave, s, d);
  const int Kout = Hh * C;
  float alpha_l = 0.0f;
  if (lane < Hh)
    alpha_l = pB[(size_t)wave * Hh + lane] / (sSum[d * Hh + lane] + 1e-16f);
  const float* hs = Hf + (size_t)s * Kout;
  float* od = outF + (size_t)d * Kout;
  for (int c = lane; c < Kout; c += 32) {
    float alpha = __shfl(alpha_l, c / C, 32);
    atomicAdd(&od[c], hs[c] * alpha);
  }
}

// ---------------- epilogue: out = leaky_relu(out + bias), in place ------------------
__global__ void bias_act(float* __restrict__ F, const float* __restrict__ b, int N, int Kout) {
  int idx = blockIdx.x * blockDim.x + threadIdx.x;
  if (idx >= N * Kout) return;
  F[idx] = lrelu(F[idx] + b[idx % Kout]);
}

// ---------------- mean pool per graph ----------------------------------------------
__global__ void pool_sum(const float* __restrict__ F, const int* __restrict__ batch,
                         float* __restrict__ gsum, float* __restrict__ gcnt, int N, int K) {
  int idx = blockIdx.x * blockDim.x + threadIdx.x;
  if (idx >= N * K) return;
  int n = idx / K, k = idx % K;
  int g = batch[n];
  atomicAdd(&gsum[g * K + k], F[idx]);
  if (k == 0) atomicAdd(&gcnt[g], 1.0f);
}
__global__ void pool_div(const float* __restrict__ gsum, const float* __restrict__ gcnt,
                         float* __restrict__ out, int G, int K) {
  int idx = blockIdx.x * blockDim.x + threadIdx.x;
  if (idx >= G * K) return;
  out[idx] = gsum[idx] / fmaxf(gcnt[idx / K], 1.0f);
}

extern "C" void kernel_launch(void* const* d_in, const int* in_sizes, int n_in,
                              void* d_out, int out_size, void* d_ws, size_t ws_size,
                              hipStream_t stream) {
  const float* x      = (const float*)d_in[0];
  const int*   ei     = (const int*)d_in[1];
  const int*   batch  = (const int*)d_in[2];
  const float* W0     = (const float*)d_in[3];
  const float* as0    = (const float*)d_in[4];
  const float* ad0    = (const float*)d_in[5];
  const float* b0     = (const float*)d_in[6];
  const float* W1     = (const float*)d_in[7];
  const float* as1    = (const float*)d_in[8];
  const float* ad1    = (const float*)d_in[9];
  const float* b1     = (const float*)d_in[10];
  const float* W2     = (const float*)d_in[11];
  const float* as2    = (const float*)d_in[12];
  const float* ad2    = (const float*)d_in[13];
  const float* b2     = (const float*)d_in[14];

  const int N    = in_sizes[0] / 128;    // 50000
  const int E    = in_sizes[1] / 2;      // 800000
  const int Etot = E + N;                // + self-loops

  // workspace layout
  float* ws = (float*)d_ws;
  size_t o = 0;
  float*    F    = ws + o; o += (size_t)N * 256;   // features / aggregation target
  float*    Hb   = ws + o; o += (size_t)N * 256;   // h = x @ W
  float*    alS  = ws + o; o += (size_t)N * 4;
  float*    alD  = ws + o; o += (size_t)N * 4;
  unsigned* mEnc = (unsigned*)(ws + o); o += (size_t)N * 4;
  float*    sSum = ws + o; o += (size_t)N * 4;
  float*    pB   = ws + o; o += (size_t)Etot * 4;
  float*    gsum = ws + o; o += 64 * 128;
  float*    gcnt = ws + o; o += 64;

  const int BLK = 256;
  auto blocks = [](long long t) { return (int)((t + 255) / 256); };

  auto layer = [&](const float* Xin, int Kin, const float* W, const float* as,
                   const float* ad, const float* bias, int Hh, int C) {
    const int Kout = Hh * C;
    // 1) h = Xin @ W  (WMMA f32)
    long long gwaves = (long long)(N / 16) * (Kout / 64);
    gemm_wmma_f32<<<blocks(gwaves * 32), BLK, 0, stream>>>(Xin, W, Hb, N, Kin, Kout);
    // 2) attention logits
    long long awaves = (long long)N * Hh;
    attn_logits<<<blocks(awaves * 32), BLK, 0, stream>>>(Hb, as, ad, alS, alD, N, Hh, C);
    // 3) init segment buffers + aggregation target (Xin no longer needed -> F reusable)
    fill_u32<<<blocks(N * Hh), BLK, 0, stream>>>(mEnc, 0x007FFFFFu /* enc(-inf) */, N * Hh);
    fill_f32<<<blocks(N * Hh), BLK, 0, stream>>>(sSum, 0.0f, N * Hh);
    fill_f32<<<blocks((long long)N * Kout), BLK, 0, stream>>>(F, 0.0f, N * Kout);
    // 4) segment softmax + weighted scatter
    edge_max<<<blocks((long long)Etot * Hh), BLK, 0, stream>>>(ei, E, Etot, alS, alD, mEnc, Hh);
    edge_exp<<<blocks((long long)Etot * Hh), BLK, 0, stream>>>(ei, E, Etot, alS, alD, mEnc, pB, sSum, Hh);
    edge_msg<<<blocks((long long)Etot * 32), BLK, 0, stream>>>(ei, E, Etot, pB, sSum, Hb, F, Hh, C);
    // 5) bias + leaky_relu (in place)
    bias_act<<<blocks((long long)N * Kout), BLK, 0, stream>>>(F, bias, N, Kout);
  };

  layer(x, 128, W0, as0, ad0, b0, 4, 64);
  layer(F, 256, W1, as1, ad1, b1, 4, 64);
  layer(F, 256, W2, as2, ad2, b2, 1, 128);   // heads=1, concat=False -> mean of 1 head

  // global mean pool over 64 graphs
  fill_f32<<<blocks(64 * 128), BLK, 0, stream>>>(gsum, 0.0f, 64 * 128);
  fill_f32<<<1, 64, 0, stream>>>(gcnt, 0.0f, 64);
  pool_sum<<<blocks((long long)N * 128), BLK, 0, stream>>>(F, batch, gsum, gcnt, N, 128);
  pool_div<<<blocks(64 * 128), BLK, 0, stream>>>(gsum, gcnt, (float*)d_out, 64, 128);
}